// SimpleGraphConv_87866440942236
// MI455X (gfx1250) — compile-verified
//
#include <hip/hip_runtime.h>
#include <hip/hip_bf16.h>

typedef __attribute__((ext_vector_type(2))) float v2f;
typedef __attribute__((ext_vector_type(8))) float v8f;

#define DIM 64
#define WT_PITCH 68  // padded LDS pitch for transposed W: stride-4 banks, conflict-free b64 loads

// ---------------------------------------------------------------------------
// copy user_emb into both output layer-0 slots (ui_list[0], soc_list[0])
// ---------------------------------------------------------------------------
__global__ void copy2_kernel(const float4* __restrict__ s, float4* __restrict__ d0,
                             float4* __restrict__ d1, long n4) {
  long i = (long)blockIdx.x * blockDim.x + threadIdx.x;
  long stride = (long)gridDim.x * blockDim.x;
  for (; i < n4; i += stride) {
    float4 v = s[i];
    d0[i] = v;
    d1[i] = v;
  }
}

// ---------------------------------------------------------------------------
// COO SpMM scatter: out[rows[e]] += val[e] * X[cols[e]]   (out pre-zeroed)
// 16 lanes per edge, each lane: float4 gather + 4 global_atomic_add_f32.
// ---------------------------------------------------------------------------
__global__ void spmm_scatter_kernel(const int* __restrict__ rows, const int* __restrict__ cols,
                                    const float* __restrict__ val, const float* __restrict__ X,
                                    float* __restrict__ out, int nedges) {
  long gid = (long)blockIdx.x * blockDim.x + threadIdx.x;
  int e = (int)(gid >> 4);
  if (e >= nedges) return;
  int part = (int)(gid & 15);
  int r = rows[e];
  int c = cols[e];
  float w = val[e];
  const float4 x = *(const float4*)(X + (size_t)c * DIM + part * 4);
  float* o = out + (size_t)r * DIM + part * 4;
  atomicAdd(o + 0, w * x.x);
  atomicAdd(o + 1, w * x.y);
  atomicAdd(o + 2, w * x.z);
  atomicAdd(o + 3, w * x.w);
}

// ---------------------------------------------------------------------------
// Fused: Y = LayerNorm(X + P@W + bias; g, beta), rows of D=64, fp32 WMMA.
// Block = 256 threads = 8 waves; each wave computes one 16x64 row tile with
// V_WMMA_F32_16X16X4_F32 (4 accumulators x 16 K-steps), then does the LN
// reduction in-register via shfl_xor butterflies within each 16-lane half.
// nrows must be a multiple of 16 (100000 and 200000 both are).
// ---------------------------------------------------------------------------
__global__ __launch_bounds__(256) void gemm_ln_kernel(
    const float* __restrict__ P, const float* __restrict__ X,
    const float* __restrict__ W, const float* __restrict__ bias,
    const float* __restrict__ g, const float* __restrict__ beta,
    float* __restrict__ Y, int nrows) {
  __shared__ float ldsWt[DIM * WT_PITCH];  // W transposed: ldsWt[c*WT_PITCH + r] = W[r][c]
  __shared__ float ldsB[DIM];
  __shared__ float ldsG[DIM];
  __shared__ float ldsBeta[DIM];

  int tid = threadIdx.x;
  for (int i = tid; i < DIM * DIM; i += 256) {
    int r = i >> 6;
    int c = i & 63;
    ldsWt[c * WT_PITCH + r] = W[i];
  }
  if (tid < DIM) {
    ldsB[tid] = bias[tid];
    ldsG[tid] = g[tid];
    ldsBeta[tid] = beta[tid];
  }
  __syncthreads();

  int wave = tid >> 5;
  int lane = tid & 31;
  int half = lane >> 4;  // 0: K pair {0,1}; 1: K pair {2,3} (A & B frag layouts)
  int l16 = lane & 15;
  int base = (blockIdx.x * 8 + wave) * 16;
  if (base >= nrows) return;  // wave-uniform: EXEC stays all-1s for WMMA

  v8f acc0 = {}, acc1 = {}, acc2 = {}, acc3 = {};
  const float* Prow = P + (size_t)(base + l16) * DIM + half * 2;
  const float* wt = &ldsWt[l16 * WT_PITCH + half * 2];

#pragma unroll
  for (int kk = 0; kk < DIM; kk += 4) {
    float2 av = *(const float2*)(Prow + kk);
    v2f a;
    a.x = av.x;
    a.y = av.y;
    float2 b0v = *(const float2*)(wt + 0 * 16 * WT_PITCH + kk);
    float2 b1v = *(const float2*)(wt + 1 * 16 * WT_PITCH + kk);
    float2 b2v = *(const float2*)(wt + 2 * 16 * WT_PITCH + kk);
    float2 b3v = *(const float2*)(wt + 3 * 16 * WT_PITCH + kk);
    v2f b0, b1, b2, b3;
    b0.x = b0v.x; b0.y = b0v.y;
    b1.x = b1v.x; b1.y = b1v.y;
    b2.x = b2v.x; b2.y = b2v.y;
    b3.x = b3v.x; b3.y = b3v.y;
    acc0 = __builtin_amdgcn_wmma_f32_16x16x4_f32(false, a, false, b0, (short)0, acc0, false, false);
    acc1 = __builtin_amdgcn_wmma_f32_16x16x4_f32(false, a, false, b1, (short)0, acc1, false, false);
    acc2 = __builtin_amdgcn_wmma_f32_16x16x4_f32(false, a, false, b2, (short)0, acc2, false, false);
    acc3 = __builtin_amdgcn_wmma_f32_16x16x4_f32(false, a, false, b3, (short)0, acc3, false, false);
  }

  // Epilogue: C/D layout -> VGPR j holds row (base + j + half*8), col = n*16 + l16.
#pragma unroll
  for (int j = 0; j < 8; ++j) {
    int row = base + j + half * 8;
    float v0 = acc0[j];
    float v1 = acc1[j];
    float v2 = acc2[j];
    float v3 = acc3[j];
    const float* xr = X + (size_t)row * DIM + l16;
    v0 += ldsB[0 + l16] + xr[0];
    v1 += ldsB[16 + l16] + xr[16];
    v2 += ldsB[32 + l16] + xr[32];
    v3 += ldsB[48 + l16] + xr[48];
    float s = v0 + v1 + v2 + v3;
    float sq = v0 * v0 + v1 * v1 + v2 * v2 + v3 * v3;
#pragma unroll
    for (int off = 1; off < 16; off <<= 1) {  // butterflies stay inside 16-lane half
      s += __shfl_xor(s, off, 32);
      sq += __shfl_xor(sq, off, 32);
    }
    float mean = s * (1.0f / 64.0f);
    float var = sq * (1.0f / 64.0f) - mean * mean;
    float inv = rsqrtf(var + 1e-5f);
    float* yr = Y + (size_t)row * DIM + l16;
    yr[0] = (v0 - mean) * inv * ldsG[0 + l16] + ldsBeta[0 + l16];
    yr[16] = (v1 - mean) * inv * ldsG[16 + l16] + ldsBeta[16 + l16];
    yr[32] = (v2 - mean) * inv * ldsG[32 + l16] + ldsBeta[32 + l16];
    yr[48] = (v3 - mean) * inv * ldsG[48 + l16] + ldsBeta[48 + l16];
  }
}

// ---------------------------------------------------------------------------
extern "C" void kernel_launch(void* const* d_in, const int* in_sizes, int n_in,
                              void* d_out, int out_size, void* d_ws, size_t ws_size,
                              hipStream_t stream) {
  const float* user_emb = (const float*)d_in[0];
  const float* item_emb = (const float*)d_in[1];
  const int* ui_src = (const int*)d_in[2];
  const int* ui_dst = (const int*)d_in[3];
  const float* ui_val = (const float*)d_in[4];
  const int* soc_src = (const int*)d_in[5];
  const int* soc_dst = (const int*)d_in[6];
  const float* soc_val = (const float*)d_in[7];
  const float* W_ui = (const float*)d_in[8];
  const float* b_ui = (const float*)d_in[9];
  const float* W_soc = (const float*)d_in[10];
  const float* b_soc = (const float*)d_in[11];
  const float* ln_g = (const float*)d_in[12];
  const float* ln_b = (const float*)d_in[13];

  const int nu = in_sizes[0] / DIM;
  const int ni = in_sizes[1] / DIM;
  const int eui = in_sizes[2];
  const int esoc = in_sizes[5];

  float* out = (float*)d_out;
  float* out_ui = out;                               // (3, nu, 64)
  float* out_soc = out + (size_t)3 * nu * DIM;       // (3, nu, 64)
  float* out_it = out_soc + (size_t)3 * nu * DIM;    // (ni, 64)

  size_t aggcap = (size_t)(nu > ni ? nu : ni) * DIM;
  float* agg = (float*)d_ws;        // shared aggregate scratch (zeroed per use)
  float* it_buf = agg + aggcap;     // item embedding after layer 0

  // layer-0 outputs are the raw user embeddings
  long n4 = (long)nu * DIM / 4;
  copy2_kernel<<<1024, 256, 0, stream>>>((const float4*)user_emb, (float4*)out_ui,
                                         (float4*)out_soc, n4);

  auto spmm = [&](const int* r, const int* c, const float* v, const float* x, float* o,
                  int rowsN, int E) {
    hipMemsetAsync(o, 0, (size_t)rowsN * DIM * sizeof(float), stream);
    long tot = (long)E * 16;
    int blocks = (int)((tot + 255) / 256);
    spmm_scatter_kernel<<<blocks, 256, 0, stream>>>(r, c, v, x, o, E);
  };
  auto gemmln = [&](const float* Pm, const float* Xm, const float* Wm, const float* bm,
                    const float* gm, const float* bet, float* Ym, int rowsN) {
    int blocks = (rowsN + 127) / 128;
    gemm_ln_kernel<<<blocks, 256, 0, stream>>>(Pm, Xm, Wm, bm, gm, bet, Ym, rowsN);
  };

  const float* u = user_emb;
  const float* it = item_emb;
  const float* s = user_emb;
  for (int l = 0; l < 2; ++l) {
    const float* Wl = W_ui + (size_t)l * DIM * DIM;
    const float* bl = b_ui + (size_t)l * DIM;
    const float* Wsl = W_soc + (size_t)l * DIM * DIM;
    const float* bsl = b_soc + (size_t)l * DIM;
    const float* g0 = ln_g + (size_t)(2 * l) * DIM;
    const float* be0 = ln_b + (size_t)(2 * l) * DIM;
    const float* g1 = ln_g + (size_t)(2 * l + 1) * DIM;
    const float* be1 = ln_b + (size_t)(2 * l + 1) * DIM;
    float* u_new = out_ui + (size_t)(l + 1) * nu * DIM;
    float* s_new = out_soc + (size_t)(l + 1) * nu * DIM;
    float* it_new = (l == 0) ? it_buf : out_it;

    // user <- items
    spmm(ui_src, ui_dst, ui_val, it, agg, nu, eui);
    gemmln(agg, u, Wl, bl, g0, be0, u_new, nu);

    // items <- users (uses previous-layer u)
    spmm(ui_dst, ui_src, ui_val, u, agg, ni, eui);
    gemmln(agg, it, Wl, bl, g1, be1, it_new, ni);

    // social: user <- user (shares LN params with user branch)
    spmm(soc_src, soc_dst, soc_val, s, agg, nu, esoc);
    gemmln(agg, s, Wsl, bsl, g0, be0, s_new, nu);

    u = u_new;
    it = it_new;
    s = s_new;
  }
}